// ComputationGraphTableParse_19069654794596
// MI455X (gfx1250) — compile-verified
//
#include <hip/hip_runtime.h>
#include <cmath>

typedef __attribute__((ext_vector_type(2))) float v2f;
typedef __attribute__((ext_vector_type(8))) float v8f;

#define NW  768   // num words
#define HID 100   // hidden size

// ---------------------------------------------------------------------------
// Generic fp32 WMMA GEMM:  out[M x Nc] = act(A[M x K] @ W[K x Nc] + bias)
// One wave (32 lanes) per 16x16 output tile, V_WMMA_F32_16X16X4_F32 over K.
// K must be a multiple of 4 (308 / 600 / 100 all are).
// ACT: 0 = identity, 1 = tanh
// ---------------------------------------------------------------------------
template <int ACT>
__global__ __launch_bounds__(32)
void wmma_gemm_f32(const float* __restrict__ A, int lda,
                   const float* __restrict__ W, int ldw,
                   const float* __restrict__ bias,
                   float* __restrict__ out, int ldo,
                   int K, int Nc) {
  const int lane = threadIdx.x;       // 0..31
  const int row  = lane & 15;
  const int hi   = lane >> 4;         // 0 or 1 (lane half)

  const int m0  = blockIdx.x * 16;
  const int n0  = blockIdx.y * 16;
  const int col = n0 + row;
  // keep EXEC uniform around WMMA: clamp address, zero via multiply
  const float cmask = (col < Nc) ? 1.0f : 0.0f;
  const int   colc  = (col < Nc) ? col : (Nc - 1);

  const float* arow = A + (size_t)(m0 + row) * lda;
  v8f acc = {};
  for (int k0 = 0; k0 < K; k0 += 4) {
    v2f a, b;
    // A 16x4: lanes 0-15 hold K = k0+0 / k0+1; lanes 16-31 hold K = k0+2 / k0+3
    a.x = arow[k0 + 2 * hi + 0];
    a.y = arow[k0 + 2 * hi + 1];
    // B 4x16: vgpr v holds K = k0 + v + 2*hi, N = col
    b.x = W[(size_t)(k0 + 2 * hi + 0) * ldw + colc] * cmask;
    b.y = W[(size_t)(k0 + 2 * hi + 1) * ldw + colc] * cmask;
    acc = __builtin_amdgcn_wmma_f32_16x16x4_f32(false, a, false, b,
                                                (short)0, acc, false, false);
  }

  if (col >= Nc) return;
  const float bv = bias ? bias[col] : 0.0f;
#pragma unroll
  for (int r = 0; r < 8; ++r) {
    float v = acc[r] + bv;
    if (ACT == 1) v = tanhf(v);
    // C/D layout: vgpr r -> row m0 + r + 8*hi, col = n0 + (lane&15)
    out[(size_t)(m0 + r + 8 * hi) * ldo + col] = v;
  }
}

// ---------------------------------------------------------------------------
// Build X[768 x 600] = concat(gather-5-neighbors(uu), hh=0), zeroed where
// indices_not_found.
// ---------------------------------------------------------------------------
__global__ __launch_bounds__(128)
void gather_kernel(const float* __restrict__ uu,
                   const int* __restrict__ indices,
                   const unsigned char* __restrict__ not_found,
                   float* __restrict__ X) {
  const int i  = blockIdx.x;
  const bool nf = not_found[i] != 0;
  const int* idx = indices + i * 5;
  float* xr = X + (size_t)i * 600;
  for (int c = threadIdx.x; c < 600; c += blockDim.x) {
    float v = 0.0f;
    if (c < 5 * HID && !nf) {
      const int j = c / HID;
      const int h = c - j * HID;
      v = uu[(size_t)idx[j] * HID + h];
    }
    xr[c] = v;
  }
}

// ---------------------------------------------------------------------------
// Pairwise head: out[i,j,c] = b2[c] + sum_k relu(hl[i,k] + hr[j,k]) * W2[k,c]
// (b1 is pre-folded into hl by the GEMM bias). 16x16 pair tile per block,
// hl/hr rows staged in LDS, 1 pair per thread.
// ---------------------------------------------------------------------------
__global__ __launch_bounds__(256)
void pairwise_kernel(const float* __restrict__ hl,
                     const float* __restrict__ hr,
                     const float* __restrict__ W2,   // (HID, 2) row-major
                     const float* __restrict__ b2,   // (2,)
                     float* __restrict__ out) {      // (NW, NW, 2)
  __shared__ float sl[16][HID];
  __shared__ float sr[16][HID];
  __shared__ float sw[HID * 2];
  __shared__ float sb[2];

  const int i0 = blockIdx.x * 16;
  const int j0 = blockIdx.y * 16;
  const int t  = threadIdx.x;

  for (int e = t; e < 16 * HID; e += 256) {
    const int r = e / HID, c = e - r * HID;
    sl[r][c] = hl[(size_t)(i0 + r) * HID + c];
    sr[r][c] = hr[(size_t)(j0 + r) * HID + c];
  }
  for (int e = t; e < 2 * HID; e += 256) sw[e] = W2[e];
  if (t < 2) sb[t] = b2[t];
  __syncthreads();

  const int ti = t >> 4;
  const int tj = t & 15;
  float a0 = sb[0], a1 = sb[1];
#pragma unroll 4
  for (int k = 0; k < HID; ++k) {
    const float s = sl[ti][k] + sr[tj][k];
    const float r = s > 0.0f ? s : 0.0f;
    a0 = fmaf(r, sw[2 * k + 0], a0);
    a1 = fmaf(r, sw[2 * k + 1], a1);
  }
  const size_t o = ((size_t)(i0 + ti) * NW + (j0 + tj)) * 2;
  out[o + 0] = a0;
  out[o + 1] = a1;
}

// ---------------------------------------------------------------------------
// Launch: uu -> gather -> bb -> oo -> (hl,hr)x3 -> pairwise x3
// ---------------------------------------------------------------------------
extern "C" void kernel_launch(void* const* d_in, const int* in_sizes, int n_in,
                              void* d_out, int out_size, void* d_ws, size_t ws_size,
                              hipStream_t stream) {
  (void)in_sizes; (void)n_in; (void)out_size; (void)ws_size;

  const float*         vv      = (const float*)d_in[0];
  const int*           indices = (const int*)d_in[1];
  const unsigned char* nf      = (const unsigned char*)d_in[2];
  // d_in[3] = num_words scalar (fixed at 768 here)
  const float* A_W   = (const float*)d_in[4];
  const float* A_b   = (const float*)d_in[5];
  const float* B_W   = (const float*)d_in[6];
  const float* B_b   = (const float*)d_in[7];
  const float* B2_Wo = (const float*)d_in[8];
  const float* B2_bo = (const float*)d_in[9];
  // d_in[10..13] = B2_Wh/bh, D_W/b: provably dead with iterations==1 outputs

  float* ws = (float*)d_ws;
  float* uu = ws;                       // 768*100
  float* X  = uu + NW * HID;            // 768*600
  float* bb = X  + NW * 600;            // 768*100
  float* oo = bb + NW * HID;            // 768*100
  float* hl = oo + NW * HID;            // 3 * 768*100
  float* hr = hl + 3 * NW * HID;        // 3 * 768*100

  const dim3 wave(32);
  const dim3 gtile(NW / 16, (HID + 15) / 16);   // 48 x 7

  // uu = tanh(vv @ A_W + A_b)            [K = 308]
  wmma_gemm_f32<1><<<gtile, wave, 0, stream>>>(vv, 308, A_W, HID, A_b, uu, HID, 308, HID);
  // X = concat(gather(uu), 0)
  gather_kernel<<<NW, 128, 0, stream>>>(uu, indices, nf, X);
  // bb = tanh(X @ B_W + B_b)             [K = 600]
  wmma_gemm_f32<1><<<gtile, wave, 0, stream>>>(X, 600, B_W, HID, B_b, bb, HID, 600, HID);
  // oo = tanh(bb @ B2_Wo + B2_bo)        [K = 100]
  wmma_gemm_f32<1><<<gtile, wave, 0, stream>>>(bb, HID, B2_Wo, HID, B2_bo, oo, HID, HID, HID);

  // per head: hl = oo @ W1[:H] + b1 (fold b1), hr = oo @ W1[H:]
  for (int h = 0; h < 3; ++h) {
    const float* W1 = (const float*)d_in[14 + 4 * h];
    const float* b1 = (const float*)d_in[15 + 4 * h];
    wmma_gemm_f32<0><<<gtile, wave, 0, stream>>>(oo, HID, W1, HID, b1,
                                                 hl + (size_t)h * NW * HID, HID, HID, HID);
    wmma_gemm_f32<0><<<gtile, wave, 0, stream>>>(oo, HID, W1 + HID * HID, HID, nullptr,
                                                 hr + (size_t)h * NW * HID, HID, HID, HID);
  }

  float* out = (float*)d_out;
  const dim3 ptile(NW / 16, NW / 16);           // 48 x 48
  for (int h = 0; h < 3; ++h) {
    const float* W2 = (const float*)d_in[16 + 4 * h];
    const float* b2 = (const float*)d_in[17 + 4 * h];
    pairwise_kernel<<<ptile, 256, 0, stream>>>(hl + (size_t)h * NW * HID,
                                               hr + (size_t)h * NW * HID,
                                               W2, b2,
                                               out + (size_t)h * NW * NW * 2);
  }
}